// LinearAttentionLayer_5153960755278
// MI455X (gfx1250) — compile-verified
//
#include <hip/hip_runtime.h>
#include <hip/hip_bf16.h>

#define B_BATCH 2
#define S_SEQ   512
#define D_DIM   1024
#define H_HEADS 8
#define HD_DIM  64
#define HID_DIM 512
#define EPSF    1e-6f
#define CHUNK   64
#define NCHUNK  (S_SEQ / CHUNK)

typedef float v2f __attribute__((ext_vector_type(2)));
typedef float v8f __attribute__((ext_vector_type(8)));

#define WMMA_F32(A_, B_, C_)                                                 \
  __builtin_amdgcn_wmma_f32_16x16x4_f32(false, (A_), false, (B_), (short)0,  \
                                        (C_), false, false)

// CDNA5 async global<->LDS DMA (16B per lane), tracked by ASYNCcnt.
// Generic LDS pointers carry the LDS byte offset in addr[31:0] (ISA 10.2).
__device__ __forceinline__ void async_b128_to_lds(void* lds_dst,
                                                  const void* gsrc) {
  asm volatile("global_load_async_to_lds_b128 %0, %1, off"
               :
               : "v"((unsigned)(unsigned long long)lds_dst), "v"(gsrc)
               : "memory");
}
__device__ __forceinline__ void async_b128_from_lds(void* gdst,
                                                    const void* lds_src) {
  asm volatile("global_store_async_from_lds_b128 %0, %1, off"
               :
               : "v"(gdst), "v"((unsigned)(unsigned long long)lds_src)
               : "memory");
}
__device__ __forceinline__ void wait_async0() {
  asm volatile("s_wait_asynccnt 0" ::: "memory");
}

// ---------------------------------------------------------------------------
// Fused QKV projection: grid.z selects {wq->Q(relu), wk->K(relu), wv->V}.
// Double-buffered async LDS staging; fp32 WMMA k-step 4; 8 waves x 16x32.
// ---------------------------------------------------------------------------
__global__ __launch_bounds__(256) void qkv_gemm_kernel(
    const float* __restrict__ A, const float* __restrict__ wq,
    const float* __restrict__ wk, const float* __restrict__ wv,
    float* __restrict__ ws_qkv, int M, int N, int K) {
  __shared__ float As[2][64][20];  // 80B rows: 16B aligned, conflict-free
  __shared__ float Bs[2][16][68];  // 272B rows

  const int z = blockIdx.z;
  const float* Bm = (z == 0) ? wq : (z == 1) ? wk : wv;
  float* C = ws_qkv + (size_t)z * M * N;
  const bool relu = (z != 2);

  const int tid  = threadIdx.x;
  const int wave = tid >> 5;
  const int lane = tid & 31;
  const int half = lane >> 4;
  const int l    = lane & 15;
  const int mBase = blockIdx.y * 64;
  const int nBase = blockIdx.x * 64;
  const int wm = (wave & 3) * 16;
  const int wn = (wave >> 2) * 32;

  const int arow = tid >> 2, acol = (tid & 3) * 4;    // A: 64x16 = 256 xb128
  const int brow = tid >> 4, bcol = (tid & 15) * 4;   // B: 16x64 = 256 xb128

  v8f acc0 = {}, acc1 = {};

  // prologue: stage tile 0 into buffer 0
  async_b128_to_lds(&As[0][arow][acol], A + (size_t)(mBase + arow) * K + acol);
  async_b128_to_lds(&Bs[0][brow][bcol], Bm + (size_t)brow * N + nBase + bcol);

  for (int k0 = 0, it = 0; k0 < K; k0 += 16, ++it) {
    const int cur = it & 1;
    wait_async0();
    __syncthreads();
    if (k0 + 16 < K) {  // stream next tile into the alternate buffer
      async_b128_to_lds(&As[cur ^ 1][arow][acol],
                        A + (size_t)(mBase + arow) * K + k0 + 16 + acol);
      async_b128_to_lds(&Bs[cur ^ 1][brow][bcol],
                        Bm + (size_t)(k0 + 16 + brow) * N + nBase + bcol);
    }
    if (k0 + 32 < K) {  // warm L2 two tiles ahead
      __builtin_prefetch(A + (size_t)(mBase + arow) * K + k0 + 32 + acol);
      __builtin_prefetch(Bm + (size_t)(k0 + 32 + brow) * N + nBase + bcol);
    }
#pragma unroll
    for (int kk = 0; kk < 16; kk += 4) {
      v2f a, b0, b1;
      a[0] = As[cur][wm + l][kk + 2 * half + 0];
      a[1] = As[cur][wm + l][kk + 2 * half + 1];
      b0[0] = Bs[cur][kk + 2 * half + 0][wn + l];
      b0[1] = Bs[cur][kk + 2 * half + 1][wn + l];
      b1[0] = Bs[cur][kk + 2 * half + 0][wn + 16 + l];
      b1[1] = Bs[cur][kk + 2 * half + 1][wn + 16 + l];
      acc0 = WMMA_F32(a, b0, acc0);
      acc1 = WMMA_F32(a, b1, acc1);
    }
  }
#pragma unroll
  for (int r = 0; r < 8; ++r) {
    int row  = mBase + wm + r + 8 * half;
    int col0 = nBase + wn + l;
    float v0 = acc0[r], v1 = acc1[r];
    if (relu) { v0 = fmaxf(v0, 0.f); v1 = fmaxf(v1, 0.f); }
    C[(size_t)row * N + col0]      = v0;
    C[(size_t)row * N + col0 + 16] = v1;
  }
}

// ---------------------------------------------------------------------------
// Output GEMM: y = XN @ wo + bo (fp32 WMMA, double-buffered async staging)
// ---------------------------------------------------------------------------
__global__ __launch_bounds__(256) void out_gemm_kernel(
    const float* __restrict__ A, const float* __restrict__ Bm,
    const float* __restrict__ bias, float* __restrict__ C,
    int M, int N, int K) {
  __shared__ float As[2][64][20];
  __shared__ float Bs[2][16][68];

  const int tid  = threadIdx.x;
  const int wave = tid >> 5;
  const int lane = tid & 31;
  const int half = lane >> 4;
  const int l    = lane & 15;
  const int mBase = blockIdx.y * 64;
  const int nBase = blockIdx.x * 64;
  const int wm = (wave & 3) * 16;
  const int wn = (wave >> 2) * 32;

  const int arow = tid >> 2, acol = (tid & 3) * 4;
  const int brow = tid >> 4, bcol = (tid & 15) * 4;

  v8f acc0 = {}, acc1 = {};

  async_b128_to_lds(&As[0][arow][acol], A + (size_t)(mBase + arow) * K + acol);
  async_b128_to_lds(&Bs[0][brow][bcol], Bm + (size_t)brow * N + nBase + bcol);

  for (int k0 = 0, it = 0; k0 < K; k0 += 16, ++it) {
    const int cur = it & 1;
    wait_async0();
    __syncthreads();
    if (k0 + 16 < K) {
      async_b128_to_lds(&As[cur ^ 1][arow][acol],
                        A + (size_t)(mBase + arow) * K + k0 + 16 + acol);
      async_b128_to_lds(&Bs[cur ^ 1][brow][bcol],
                        Bm + (size_t)(k0 + 16 + brow) * N + nBase + bcol);
    }
    if (k0 + 32 < K) {
      __builtin_prefetch(A + (size_t)(mBase + arow) * K + k0 + 32 + acol);
      __builtin_prefetch(Bm + (size_t)(k0 + 32 + brow) * N + nBase + bcol);
    }
#pragma unroll
    for (int kk = 0; kk < 16; kk += 4) {
      v2f a, b0, b1;
      a[0] = As[cur][wm + l][kk + 2 * half + 0];
      a[1] = As[cur][wm + l][kk + 2 * half + 1];
      b0[0] = Bs[cur][kk + 2 * half + 0][wn + l];
      b0[1] = Bs[cur][kk + 2 * half + 1][wn + l];
      b1[0] = Bs[cur][kk + 2 * half + 0][wn + 16 + l];
      b1[1] = Bs[cur][kk + 2 * half + 1][wn + 16 + l];
      acc0 = WMMA_F32(a, b0, acc0);
      acc1 = WMMA_F32(a, b1, acc1);
    }
  }
#pragma unroll
  for (int r = 0; r < 8; ++r) {
    int row  = mBase + wm + r + 8 * half;
    int col0 = nBase + wn + l;
    C[(size_t)row * N + col0]      = acc0[r] + bias[col0];
    C[(size_t)row * N + col0 + 16] = acc1[r] + bias[col0 + 16];
  }
}

// ---------------------------------------------------------------------------
// Chunked reset-scan per (b,h), chunk C=64, all-WMMA, double-buffered chunks.
// rsAll[c][s] = #resets in chunk c positions [0..s]; B_t survives at s iff
// rs[s]==rs[t] (t<=s); incoming state survives iff rs[s]==0.
// ---------------------------------------------------------------------------
__global__ __launch_bounds__(256) void chunk_scan_kernel(
    const float* __restrict__ Q, const float* __restrict__ Kp,
    const float* __restrict__ Vp, const int* __restrict__ mask,
    const float* __restrict__ carry, float* __restrict__ X,
    float* __restrict__ carry_out) {
  __shared__ float Qs[2][64][68];
  __shared__ float Ks[2][64][68];
  __shared__ float Vs[2][64][68];
  __shared__ float Ps[64][68];
  __shared__ float Ss[64][68];
  __shared__ int   mskAll[S_SEQ];
  __shared__ int   rsAll[S_SEQ];

  const int bh = blockIdx.x, b = bh >> 3, h = bh & 7;
  const int tid  = threadIdx.x;
  const int wave = tid >> 5;
  const int lane = tid & 31;
  const int half = lane >> 4;
  const int l    = lane & 15;
  const int mrow = (wave & 3) * 16;
  const int nb0  = (wave >> 2) * 32;
  const size_t headOff = (size_t)h * HD_DIM;

  // prologue: async-stage chunk 0 + carry, and precompute all reset prefixes
#pragma unroll
  for (int r = 0; r < 4; ++r) {
    int idx = tid + 256 * r;
    int t = idx >> 4, e4 = (idx & 15) * 4;
    size_t g = ((size_t)(b * S_SEQ + t)) * HID_DIM + headOff + e4;
    async_b128_to_lds(&Qs[0][t][e4], Q + g);
    async_b128_to_lds(&Ks[0][t][e4], Kp + g);
    async_b128_to_lds(&Vs[0][t][e4], Vp + g);
    async_b128_to_lds(&Ss[t][e4], carry + (size_t)bh * 4096 + (size_t)idx * 4);
  }
#pragma unroll
  for (int r = 0; r < 2; ++r) {
    int idx = tid + 256 * r;
    mskAll[idx] = mask[b * S_SEQ + idx];
  }
  __syncthreads();
#pragma unroll
  for (int r = 0; r < 2; ++r) {
    int j = tid + 256 * r;
    int base = j & ~63, t = j & 63;
    int acc = 0;
    for (int u = 0; u <= t; ++u) acc += mskAll[base + u];
    rsAll[j] = acc;
  }
  __syncthreads();

  for (int c = 0; c < NCHUNK; ++c) {
    const int cur = c & 1;
    const int s0 = c * CHUNK;
    const int* rsc = &rsAll[s0];

    wait_async0();       // chunk c tiles (and carry on c==0) are in LDS
    __syncthreads();
    if (c + 1 < NCHUNK) {  // stream chunk c+1 into the alternate buffers
#pragma unroll
      for (int r = 0; r < 4; ++r) {
        int idx = tid + 256 * r;
        int t = idx >> 4, e4 = (idx & 15) * 4;
        size_t g = ((size_t)(b * S_SEQ + s0 + CHUNK + t)) * HID_DIM + headOff + e4;
        async_b128_to_lds(&Qs[cur ^ 1][t][e4], Q + g);
        async_b128_to_lds(&Ks[cur ^ 1][t][e4], Kp + g);
        async_b128_to_lds(&Vs[cur ^ 1][t][e4], Vp + g);
      }
    }

    // ---- Phase 1: P = Q K^T, masked by causal/segment ----
    {
      v8f p0 = {}, p1 = {};
#pragma unroll
      for (int kk = 0; kk < 64; kk += 4) {
        v2f a, b0, b1;
        a[0]  = Qs[cur][mrow + l][kk + 2 * half + 0];
        a[1]  = Qs[cur][mrow + l][kk + 2 * half + 1];
        b0[0] = Ks[cur][nb0 + l][kk + 2 * half + 0];
        b0[1] = Ks[cur][nb0 + l][kk + 2 * half + 1];
        b1[0] = Ks[cur][nb0 + 16 + l][kk + 2 * half + 0];
        b1[1] = Ks[cur][nb0 + 16 + l][kk + 2 * half + 1];
        p0 = WMMA_F32(a, b0, p0);
        p1 = WMMA_F32(a, b1, p1);
      }
#pragma unroll
      for (int r = 0; r < 8; ++r) {
        int s = mrow + r + 8 * half;
        int t0 = nb0 + l, t1 = t0 + 16;
        Ps[s][t0] = (t0 <= s && rsc[s] == rsc[t0]) ? p0[r] : 0.f;
        Ps[s][t1] = (t1 <= s && rsc[s] == rsc[t1]) ? p1[r] : 0.f;
      }
    }
    __syncthreads();

    // ---- Phase 2: X = keep0 .* (Q S_in^T) + P' V -> global ----
    {
      v8f x0 = {}, x1 = {};
#pragma unroll
      for (int kk = 0; kk < 64; kk += 4) {
        v2f a, b0, b1;
        a[0]  = Qs[cur][mrow + l][kk + 2 * half + 0];
        a[1]  = Qs[cur][mrow + l][kk + 2 * half + 1];
        b0[0] = Ss[nb0 + l][kk + 2 * half + 0];
        b0[1] = Ss[nb0 + l][kk + 2 * half + 1];
        b1[0] = Ss[nb0 + 16 + l][kk + 2 * half + 0];
        b1[1] = Ss[nb0 + 16 + l][kk + 2 * half + 1];
        x0 = WMMA_F32(a, b0, x0);
        x1 = WMMA_F32(a, b1, x1);
      }
#pragma unroll
      for (int r = 0; r < 8; ++r) {
        float f = (rsc[mrow + r + 8 * half] == 0) ? 1.f : 0.f;
        x0[r] *= f;
        x1[r] *= f;
      }
#pragma unroll
      for (int kk = 0; kk < 64; kk += 4) {
        v2f a, b0, b1;
        a[0]  = Ps[mrow + l][kk + 2 * half + 0];
        a[1]  = Ps[mrow + l][kk + 2 * half + 1];
        b0[0] = Vs[cur][kk + 2 * half + 0][nb0 + l];
        b0[1] = Vs[cur][kk + 2 * half + 1][nb0 + l];
        b1[0] = Vs[cur][kk + 2 * half + 0][nb0 + 16 + l];
        b1[1] = Vs[cur][kk + 2 * half + 1][nb0 + 16 + l];
        x0 = WMMA_F32(a, b0, x0);
        x1 = WMMA_F32(a, b1, x1);
      }
#pragma unroll
      for (int r = 0; r < 8; ++r) {
        int s = mrow + r + 8 * half;
        size_t g = ((size_t)(b * S_SEQ + s0 + s)) * HID_DIM + headOff + nb0 + l;
        X[g]      = x0[r];
        X[g + 16] = x1[r];
      }
    }
    __syncthreads();

    // ---- Phase 3: S = [rEnd==0] S + V^T diag(rs[t]==rEnd) K ----
    {
      const int rEnd = rsc[63];
      const float fIn = (rEnd == 0) ? 1.f : 0.f;
      v8f s0v, s1v;
#pragma unroll
      for (int r = 0; r < 8; ++r) {
        int i = mrow + r + 8 * half;
        s0v[r] = fIn * Ss[i][nb0 + l];
        s1v[r] = fIn * Ss[i][nb0 + 16 + l];
      }
#pragma unroll
      for (int kk = 0; kk < 64; kk += 4) {
        int t0 = kk + 2 * half, t1 = t0 + 1;
        float g0 = (rsc[t0] == rEnd) ? 1.f : 0.f;
        float g1 = (rsc[t1] == rEnd) ? 1.f : 0.f;
        v2f a, b0, b1;
        a[0]  = Vs[cur][t0][mrow + l];
        a[1]  = Vs[cur][t1][mrow + l];
        b0[0] = g0 * Ks[cur][t0][nb0 + l];
        b0[1] = g1 * Ks[cur][t1][nb0 + l];
        b1[0] = g0 * Ks[cur][t0][nb0 + 16 + l];
        b1[1] = g1 * Ks[cur][t1][nb0 + 16 + l];
        s0v = WMMA_F32(a, b0, s0v);
        s1v = WMMA_F32(a, b1, s1v);
      }
#pragma unroll
      for (int r = 0; r < 8; ++r) {  // each wave rewrites only its own tile
        int i = mrow + r + 8 * half;
        Ss[i][nb0 + l]      = s0v[r];
        Ss[i][nb0 + 16 + l] = s1v[r];
      }
    }
    __syncthreads();
  }

  // final carry: LDS -> global via async DMA store
#pragma unroll
  for (int r = 0; r < 4; ++r) {
    int idx = tid + 256 * r;
    int t = idx >> 4, e4 = (idx & 15) * 4;
    async_b128_from_lds(carry_out + (size_t)bh * 4096 + (size_t)idx * 4,
                        &Ss[t][e4]);
  }
  wait_async0();
}

// ---------------------------------------------------------------------------
// RMSNorm over HID=512 per (b,s) row (float4)
// ---------------------------------------------------------------------------
__global__ __launch_bounds__(128) void rmsnorm_kernel(
    const float* __restrict__ X, const float* __restrict__ scale,
    float* __restrict__ XN) {
  __shared__ float red[4];
  const int row = blockIdx.x;
  const int tid = threadIdx.x;
  const float4 v = ((const float4*)(X + (size_t)row * HID_DIM))[tid];
  float ssum = v.x * v.x + v.y * v.y + v.z * v.z + v.w * v.w;
#pragma unroll
  for (int off = 16; off > 0; off >>= 1) ssum += __shfl_down(ssum, off, 32);
  if ((tid & 31) == 0) red[tid >> 5] = ssum;
  __syncthreads();
  const float inv =
      rsqrtf((red[0] + red[1] + red[2] + red[3]) * (1.0f / HID_DIM) + EPSF);
  const float4 sc = ((const float4*)scale)[tid];
  float4 o;
  o.x = v.x * inv * sc.x;
  o.y = v.y * inv * sc.y;
  o.z = v.z * inv * sc.z;
  o.w = v.w * inv * sc.w;
  ((float4*)(XN + (size_t)row * HID_DIM))[tid] = o;
}

// ---------------------------------------------------------------------------
extern "C" void kernel_launch(void* const* d_in, const int* in_sizes, int n_in,
                              void* d_out, int out_size, void* d_ws,
                              size_t ws_size, hipStream_t stream) {
  (void)in_sizes; (void)n_in; (void)out_size; (void)ws_size;
  const float* inputs    = (const float*)d_in[0];
  const int*   mask      = (const int*)d_in[1];
  const float* carry     = (const float*)d_in[2];
  const float* wq        = (const float*)d_in[3];
  const float* wk        = (const float*)d_in[4];
  const float* wv        = (const float*)d_in[5];
  const float* rms_scale = (const float*)d_in[6];
  const float* wo        = (const float*)d_in[7];
  const float* bo        = (const float*)d_in[8];

  float* out       = (float*)d_out;
  float* carry_out = out;  // 65536 floats
  float* y         = out + (size_t)B_BATCH * H_HEADS * HD_DIM * HD_DIM;

  float* ws = (float*)d_ws;
  const size_t QKV = (size_t)B_BATCH * S_SEQ * HID_DIM;  // 524288 floats
  float* Q  = ws + 0 * QKV;
  float* Kp = ws + 1 * QKV;
  float* V  = ws + 2 * QKV;
  float* X  = ws + 3 * QKV;
  float* XN = ws + 4 * QKV;

  const int M = B_BATCH * S_SEQ;  // 1024
  dim3 blk(256);

  dim3 gQKV(HID_DIM / 64, M / 64, 3);  // (8,16,3)
  qkv_gemm_kernel<<<gQKV, blk, 0, stream>>>(inputs, wq, wk, wv, Q,
                                            M, HID_DIM, D_DIM);

  chunk_scan_kernel<<<dim3(B_BATCH * H_HEADS), blk, 0, stream>>>(
      Q, Kp, V, mask, carry, X, carry_out);

  rmsnorm_kernel<<<dim3(M), dim3(128), 0, stream>>>(X, rms_scale, XN);

  dim3 gOut(D_DIM / 64, M / 64);  // (16,16)
  out_gemm_kernel<<<gOut, blk, 0, stream>>>(XN, wo, bo, y, M, D_DIM, HID_DIM);
}